// SKA_Small_30545807409744
// MI455X (gfx1250) — compile-verified
//
#include <hip/hip_runtime.h>
#include <stdint.h>

// SKA_Small: per-pixel dynamic 3x3 aggregation, weights shared across the 8
// channels of each group. Memory-bound (AI ~1.4 F/B) -> optimize the data path:
//   - TENSOR_LOAD_TO_LDS (TDM) DMA staging of the x tile: each of the 8 waves
//     issues one 2D-tile descriptor (its channel) -> DMA does all addr math
//   - per-pixel weights streamed (non-temporal) WHILE the DMA is in flight
//   - s_wait_tensorcnt 0 (own channel) + workgroup barrier releases consumers
//   - conflict-free LDS layout, 9x spatial reuse of x served from LDS

#define HH 128
#define WW 128
#define HWELEMS (HH * WW)
#define TH 8  // output rows per block

typedef uint32_t u32x4 __attribute__((ext_vector_type(4)));
typedef uint32_t u32x8 __attribute__((ext_vector_type(8)));

__device__ __forceinline__ void tdm_load_2d(u32x4 g0, u32x8 g1) {
  // TENSOR_LOAD_TO_LDS, 2D tile: VADDR0 = D# group0 (4 SGPRs),
  // VADDR1 = D# group1 (8 SGPRs); groups 2/3 omitted (<=2D tensor).
  asm volatile("tensor_load_to_lds %0, %1"
               :: "s"(g0), "s"(g1)
               : "memory");
}

__device__ __forceinline__ void wait_tensor0() {
#if __has_builtin(__builtin_amdgcn_s_wait_tensorcnt)
  __builtin_amdgcn_s_wait_tensorcnt(0);
#else
  asm volatile("s_wait_tensorcnt 0" ::: "memory");
#endif
}

__launch_bounds__(256)
__global__ void ska_small_kernel(const float* __restrict__ x,
                                 const float* __restrict__ w,
                                 float* __restrict__ out) {
  // 8 channels x 10 rows (8 + 1-row halo top/bottom) x 128 cols = 40 KB LDS
  __shared__ __align__(16) float xs[8 * 10 * WW];

  const int tid  = threadIdx.x;
  const int bid  = blockIdx.x;
  const int tile = bid & 15;         // H/TH = 16 row-tiles
  const int g    = (bid >> 4) & 7;   // group
  const int b    = bid >> 7;         // batch
  const int h0   = tile * TH;

  // Row clamp for the tile halo (all scalar): interior tiles load 10 rows,
  // edge tiles load 9 and zero-fill the missing halo row in LDS.
  const int rows_lo = (h0 == 0) ? 0 : h0 - 1;
  const int rows_hi = (h0 + TH < HH) ? h0 + TH : HH - 1;
  const int nrows   = rows_hi - rows_lo + 1;      // 9 or 10
  const int rowoff  = rows_lo - (h0 - 1);         // 0 or 1 (LDS row offset)

  const float* xg = x + (size_t)(b * 64 + g * 8) * HWELEMS;

  // ---- Stage x tile via the Tensor Data Mover: one descriptor per wave ----
  // Wave i DMAs channel i's 128-col x nrows tile into its 10-row LDS slot.
  {
    const int c = __builtin_amdgcn_readfirstlane(tid >> 5);  // wave-uniform
    const uint32_t lds_base =
        (uint32_t)__builtin_amdgcn_readfirstlane((int)(uint32_t)(uintptr_t)&xs[0]);
    const uint64_t ga =
        (uint64_t)(uintptr_t)(xg + ((size_t)c * HH + rows_lo) * WW);
    // D# group0: count=1 | lds_addr | global_addr | type=2 (image)
    u32x4 g0 = {
        1u,
        lds_base + (uint32_t)((c * 10 + rowoff) * (WW * 4)),
        (uint32_t)ga,
        (uint32_t)(ga >> 32) | 0x80000000u
    };
    // D# group1: data_size=4B(code 2); tensor_dim0=128, tensor_dim1=nrows;
    // tile_dim0=128, tile_dim1=nrows; tensor_dim0_stride=128 (row stride).
    u32x8 g1 = {
        0x00020000u,                 // data_size=2<<16, mask/flags=0
        (uint32_t)WW << 16,          // tensor_dim0[15:0] in bits 63:48
        (uint32_t)nrows << 16,       // tensor_dim1[15:0] in bits 95:80
        (uint32_t)WW << 16,          // tile_dim0 in bits 127:112
        (uint32_t)nrows,             // tile_dim1 in bits 143:128
        (uint32_t)WW,                // tensor_dim0_stride[31:0]
        0u,                          // stride0[47:32] | stride1[15:0]
        0u                           // stride1[47:16]
    };
    tdm_load_2d(g0, g1);
  }

  // Zero-fill the one halo row TDM skipped on edge tiles (block-uniform;
  // touches LDS rows the DMA does not write).
  if (nrows != 10) {
    const int rr = (h0 == 0) ? 0 : 9;
    const int c  = tid >> 5;          // 0..7
    const int u  = tid & 31;          // 16B unit
    float* lp = &xs[(c * 10 + rr) * WW + u * 4];
    lp[0] = 0.f; lp[1] = 0.f; lp[2] = 0.f; lp[3] = 0.f;
  }

  // ---- Stream the 36 per-pixel weights while the tensor DMA is in flight ---
  const int col  = tid & (WW - 1);
  const int half = tid >> 7;               // 0: rows 0..3, 1: rows 4..7
  const bool cL  = (col == 0);
  const bool cR  = (col == WW - 1);
  const int cm1  = cL ? col : col - 1;     // clamped (tap weight zeroed instead)
  const int cp1  = cR ? col : col + 1;

  const float* wb = w   + (size_t)(b * 8 + g) * 9 * HWELEMS;
  float*       ob = out + (size_t)(b * 64 + g * 8) * HWELEMS;

  float wk[4][9];
#pragma unroll
  for (int r4 = 0; r4 < 4; ++r4) {
    const int gh = h0 + half * 4 + r4;
#pragma unroll
    for (int k = 0; k < 9; ++k)
      wk[r4][k] =
          __builtin_nontemporal_load(wb + (size_t)k * HWELEMS + gh * WW + col);
    if (cL) { wk[r4][0] = 0.f; wk[r4][3] = 0.f; wk[r4][6] = 0.f; }  // kj==0 off
    if (cR) { wk[r4][2] = 0.f; wk[r4][5] = 0.f; wk[r4][8] = 0.f; }  // kj==2 off
  }

  wait_tensor0();      // s_wait_tensorcnt 0: this wave's channel landed
  __syncthreads();     // all 8 channels visible (also drains DScnt for pads)

  // ---- Compute: 1 column x 4 rows per thread, all 8 channels ----
#pragma unroll
  for (int r4 = 0; r4 < 4; ++r4) {
    const int r  = half * 4 + r4;
    const int gh = h0 + r;
#pragma unroll
    for (int c = 0; c < 8; ++c) {
      // LDS rows r, r+1, r+2 hold global rows gh-1, gh, gh+1 for channel c.
      const float* x0 = &xs[(c * 10 + r) * WW];
      float acc =      x0[cm1]           * wk[r4][0];
      acc = fmaf(x0[col],            wk[r4][1], acc);
      acc = fmaf(x0[cp1],            wk[r4][2], acc);
      acc = fmaf(x0[WW + cm1],       wk[r4][3], acc);
      acc = fmaf(x0[WW + col],       wk[r4][4], acc);
      acc = fmaf(x0[WW + cp1],       wk[r4][5], acc);
      acc = fmaf(x0[2 * WW + cm1],   wk[r4][6], acc);
      acc = fmaf(x0[2 * WW + col],   wk[r4][7], acc);
      acc = fmaf(x0[2 * WW + cp1],   wk[r4][8], acc);
      __builtin_nontemporal_store(acc, ob + ((size_t)c * HH + gh) * WW + col);
    }
  }
}

extern "C" void kernel_launch(void* const* d_in, const int* in_sizes, int n_in,
                              void* d_out, int out_size, void* d_ws, size_t ws_size,
                              hipStream_t stream) {
  (void)in_sizes; (void)n_in; (void)out_size; (void)d_ws; (void)ws_size;
  const float* x = (const float*)d_in[0];  // (8, 64, 128, 128) fp32
  const float* w = (const float*)d_in[1];  // (8, 8, 1, 9, 128, 128) fp32
  float* out = (float*)d_out;              // (8, 64, 128, 128) fp32

  const int blocks = 8 * 8 * (HH / TH);    // B * G * row-tiles = 1024
  ska_small_kernel<<<blocks, 256, 0, stream>>>(x, w, out);
}